// DecodeDetections_20169166422178
// MI455X (gfx1250) — compile-verified
//
#include <hip/hip_runtime.h>
#include <stdint.h>

// ---------------- problem constants (match reference) ----------------
#define B_       32
#define N_       8732      // anchors
#define NCLS_    20        // foreground classes
#define CH_      33        // NCLS + 13 channels per anchor
#define TOPK_    200
#define NMSMAX_  100
#define CONF_T_  0.01f
#define IOU_T_   0.45f
#define IMG_     512.0f

#define NMS_THREADS 512                  // 16 wave32s; 1 WG/WGP at this LDS size
#define NMS_WAVES   (NMS_THREADS / 32)
#define BOXB     (N_ * 16)               // 139712 B : float4 boxes in LDS
#define SCB      (N_ * 4)                // 34928  B : scores in LDS
#define REDOFF   (BOXB + SCB)            // 174640 B : reduction scratch
#define NMS_LDS  (REDOFF + (NMS_WAVES + 1) * 8 + 8)  // < 320KB/WGP

static_assert(N_ % 4 == 0, "vectorized score path assumes N % 4 == 0");

typedef uint32_t v4u __attribute__((ext_vector_type(4)));
typedef int      v4i __attribute__((ext_vector_type(4)));
typedef int      v8i __attribute__((ext_vector_type(8)));

// ---- CDNA5 async global->LDS copy (ASYNCcnt path, ISA ch.10/15.18.3) ----
__device__ __forceinline__ void async_ld_b128(uint32_t lds_addr, uint32_t voff,
                                              const void* sbase) {
  asm volatile("global_load_async_to_lds_b128 %0, %1, %2"
               :: "v"(lds_addr), "v"(voff), "s"(sbase)
               : "memory");
}
__device__ __forceinline__ void wait_async0() {
  asm volatile("s_wait_asynccnt 0" ::: "memory");
}

// ---- CDNA5 Tensor Data Mover: 1-D tile DMA global->LDS (TENSORcnt path) ----
// nelem8: tile length in 8-byte units (data_size=3). Must be issued by a wave
// that reaches the instruction (TDM ignores EXEC), so call under a
// wave-uniform branch only.
__device__ __forceinline__ void tdm_load_1d(uint32_t lds_addr, const void* gptr,
                                            uint32_t nelem8) {
  uint64_t ga = (uint64_t)(uintptr_t)gptr;
  v4u g0;
  g0.x = 1u;                                        // count=1, user mode
  g0.y = lds_addr;                                  // LDS byte address
  g0.z = (uint32_t)(ga & 0xffffffffu);              // global_addr[31:0]
  g0.w = (uint32_t)((ga >> 32) & 0x01ffffffu)       // global_addr[56:32]
         | (2u << 30);                              // type = 2 ("image")
  v8i g1;
  g1[0] = (int)(3u << 16);                          // data_size = 8B; mask=0
  g1[1] = (int)((nelem8 & 0xffffu) << 16);          // tensor_dim0[15:0]
  g1[2] = (int)(((nelem8 >> 16) & 0xffffu)          // tensor_dim0[31:16]
                | (1u << 16));                      // tensor_dim1 = 1
  g1[3] = (int)((nelem8 & 0xffffu) << 16);          // tile_dim0 = nelem8
  g1[4] = 1;                                        // tile_dim1 = 1
  g1[5] = (int)nelem8;                              // tensor_dim0_stride
  g1[6] = 0;
  g1[7] = 0;
  v4i gz = {0, 0, 0, 0};
#if defined(__clang_major__) && (__clang_major__ >= 23)
  v8i gz8 = {0, 0, 0, 0, 0, 0, 0, 0};
  __builtin_amdgcn_tensor_load_to_lds(g0, g1, gz, gz, gz8, 0);
#else
  __builtin_amdgcn_tensor_load_to_lds(g0, g1, gz, gz, 0);
#endif
}

// float -> order-preserving uint32 (monotonic for all finite + inf values)
__device__ __forceinline__ uint32_t f2ord(float v) {
  uint32_t u = __float_as_uint(v);
  return (u & 0x80000000u) ? ~u : (u | 0x80000000u);
}
#define ORD_NEGINF 0x007FFFFFu   // f2ord(-inf)

// ---------------------------------------------------------------------
// Kernel 1: decode boxes + threshold scores into class-major layout.
//   boxes  : [B][N] float4 (xmin,ymin,xmax,ymax) * 512
//   scoresT: [B][C][N] float, -inf where score <= CONF_T
// ---------------------------------------------------------------------
__global__ __launch_bounds__(256) void decode_kernel(
    const float* __restrict__ y, float4* __restrict__ boxes,
    float* __restrict__ scoresT) {
  int i = blockIdx.x * blockDim.x + threadIdx.x;   // i = b*N + n
  if (i >= B_ * N_) return;
  int b = i / N_;
  int n = i - b * N_;
  const float* r = y + (size_t)i * CH_;

  float cx_p = r[21], cy_p = r[22], w_p = r[23], h_p = r[24];
  float x1a  = r[25], y1a  = r[26], x2a = r[27], y2a = r[28];
  float vcx  = r[29], vcy  = r[30], vw  = r[31], vh  = r[32];

  float wa  = x2a - x1a, ha = y2a - y1a;
  float cxa = (x2a + x1a) * 0.5f, cya = (y2a + y1a) * 0.5f;
  float cx  = cx_p * vcx * wa + cxa;
  float cy  = cy_p * vcy * ha + cya;
  float w   = expf(w_p * vw) * wa;
  float h   = expf(h_p * vh) * ha;

  float4 box;
  box.x = (cx - 0.5f * w) * IMG_;
  box.y = (cy - 0.5f * h) * IMG_;
  box.z = (cx + 0.5f * w) * IMG_;
  box.w = (cy + 0.5f * h) * IMG_;
  boxes[i] = box;

#pragma unroll
  for (int c = 0; c < NCLS_; ++c) {
    float s = r[1 + c];
    scoresT[((size_t)b * NCLS_ + c) * N_ + n] =
        (s > CONF_T_) ? s : -__builtin_inff();
  }
}

// ---------------------------------------------------------------------
// Kernel 2: greedy NMS, one workgroup per (b,c). Fully LDS-resident
// (175KB of the 320KB WGP LDS). Boxes staged by the TDM, scores by
// per-lane async-to-LDS. Suppression writes -inf into the LDS score,
// so "available" == score > -inf exactly as in the reference.
// ---------------------------------------------------------------------
__global__ __launch_bounds__(NMS_THREADS) void nms_kernel(
    const float4* __restrict__ boxes, const float* __restrict__ scoresT,
    float* __restrict__ sel_score, int* __restrict__ sel_idx) {
  extern __shared__ unsigned char smem[];
  float4* bx = (float4*)smem;                              // [N_]
  float4* sv = (float4*)(smem + BOXB);                     // [N_/4]
  float*  s  = (float*)(smem + BOXB);                      // [N_]
  unsigned long long* red = (unsigned long long*)(smem + REDOFF);

  const int bc  = blockIdx.x;           // b*NCLS_ + c
  const int b   = bc / NCLS_;
  const int tid = threadIdx.x;

  const uint32_t dynbase = __builtin_amdgcn_groupstaticsize();
  const float4* gb = boxes + (size_t)b * N_;               // 16B aligned
  const float*  gs = scoresT + (size_t)bc * N_;            // 16B aligned

  // ---- boxes tile (139.7KB) via Tensor Data Mover: wave 0 only (TDM
  //      ignores EXEC, so keep the branch wave-uniform) ----
  if (tid < 32) {
    tdm_load_1d(dynbase, gb, (uint32_t)(N_ * 2) /* 8-byte units */);
    __builtin_amdgcn_s_wait_tensorcnt(0);
  }
  // ---- scores tile (34.9KB) via per-lane async-to-LDS ----
  for (int i = tid; i < N_ / 4; i += NMS_THREADS)          // 2183 x b128
    async_ld_b128(dynbase + BOXB + (uint32_t)(i * 16), (uint32_t)(i * 16), gs);
  wait_async0();
  __syncthreads();

  const int out_base = bc * NMSMAX_;
  int it = 0;
  for (; it < NMSMAX_; ++it) {
    // ---- argmax over LDS scores, 4 per ds_load_b128;
    //      key = (ord(score), ~n) -> max key picks the highest score,
    //      ties to the smallest anchor index (matches stable argsort) ----
    unsigned long long best = ((unsigned long long)ORD_NEGINF << 32);
    for (int j4 = tid; j4 < N_ / 4; j4 += NMS_THREADS) {
      float4 v = sv[j4];
      uint32_t base = (uint32_t)(j4 * 4);
#pragma unroll
      for (int k = 0; k < 4; ++k) {
        float vk = (k == 0) ? v.x : (k == 1) ? v.y : (k == 2) ? v.z : v.w;
        unsigned long long key =
            ((unsigned long long)f2ord(vk) << 32) | (uint32_t)(~(base + k));
        best = (key > best) ? key : best;
      }
    }
#pragma unroll
    for (int off = 16; off > 0; off >>= 1) {               // wave32 reduce
      unsigned long long o = __shfl_xor(best, off, 32);
      best = (o > best) ? o : best;
    }
    if ((tid & 31) == 0) red[tid >> 5] = best;
    __syncthreads();
    if (tid == 0) {
      unsigned long long m = red[0];
#pragma unroll
      for (int w = 1; w < NMS_WAVES; ++w)
        m = (red[w] > m) ? red[w] : m;
      red[NMS_WAVES] = m;
    }
    __syncthreads();
    best = red[NMS_WAVES];

    uint32_t ord = (uint32_t)(best >> 32);
    if (ord == ORD_NEGINF) break;                          // nothing left
    int n = (int)(~(uint32_t)(best & 0xffffffffu));

    if (tid == 0) {
      sel_idx[out_base + it]   = n;
      // thresholded scores are always positive => bits = ord & 0x7fffffff
      sel_score[out_base + it] = __uint_as_float(ord & 0x7FFFFFFFu);
    }

    // ---- suppression: IoU(picked, j) > thr  (reference's +1e-9 denom) ----
    float4 bi = bx[n];                                      // LDS broadcast
    float areai = fmaxf(bi.z - bi.x, 0.f) * fmaxf(bi.w - bi.y, 0.f);
    for (int j = tid; j < N_; j += NMS_THREADS) {
      float4 bj = bx[j];
      float ix1 = fmaxf(bi.x, bj.x), iy1 = fmaxf(bi.y, bj.y);
      float ix2 = fminf(bi.z, bj.z), iy2 = fminf(bi.w, bj.w);
      float inter = fmaxf(ix2 - ix1, 0.f) * fmaxf(iy2 - iy1, 0.f);
      float areaj = fmaxf(bj.z - bj.x, 0.f) * fmaxf(bj.w - bj.y, 0.f);
      float iou = inter / (areai + areaj - inter + 1e-9f);
      if (iou > IOU_T_ || j == n) s[j] = -__builtin_inff();
    }
    __syncthreads();
  }
  // remaining slots invalid (their output is fully masked downstream)
  for (int k = it + tid; k < NMSMAX_; k += NMS_THREADS) {
    sel_score[out_base + k] = -__builtin_inff();
    sel_idx[out_base + k]   = 0;
  }
}

// ---------------------------------------------------------------------
// Kernel 3: per-batch top-200 over C*NMS_MAX=2000 candidates via a 2048
// bitonic sort on 64-bit keys (score desc, flat index asc), then emit
// [cls, score, xmin, ymin, xmax, ymax]; invalid -> [1,0,0,0,0,0].
// ---------------------------------------------------------------------
#define M2_ (NCLS_ * NMSMAX_)   // 2000
__global__ __launch_bounds__(256) void topk_kernel(
    const float* __restrict__ sel_score, const int* __restrict__ sel_idx,
    const float4* __restrict__ boxes, float* __restrict__ out) {
  __shared__ unsigned long long keys[2048];
  const int b = blockIdx.x, tid = threadIdx.x;

  for (int f = tid; f < 2048; f += 256) {
    uint32_t ord = (f < M2_) ? f2ord(sel_score[b * M2_ + f]) : ORD_NEGINF;
    keys[f] = ((unsigned long long)ord << 32) | (uint32_t)(~(uint32_t)f);
  }
  __syncthreads();

  for (int k = 2; k <= 2048; k <<= 1) {
    for (int j = k >> 1; j > 0; j >>= 1) {
      for (int t = tid; t < 1024; t += 256) {
        int i = 2 * t - (t & (j - 1));     // all i with (i & j) == 0
        int m = i | j;
        bool dirDesc = ((i & k) == 0);
        unsigned long long a = keys[i], c = keys[m];
        bool sw = dirDesc ? (a < c) : (a > c);
        if (sw) { keys[i] = c; keys[m] = a; }
      }
      __syncthreads();
    }
  }

  for (int k = tid; k < TOPK_; k += 256) {
    unsigned long long key = keys[k];
    uint32_t ord = (uint32_t)(key >> 32);
    float* o = out + ((size_t)b * TOPK_ + k) * 6;
    if (ord == ORD_NEGINF) {               // invalid: where(valid,..)+1 path
      o[0] = 1.f; o[1] = 0.f; o[2] = 0.f; o[3] = 0.f; o[4] = 0.f; o[5] = 0.f;
    } else {
      int f = (int)(~(uint32_t)(key & 0xffffffffu));
      int c = f / NMSMAX_;
      int n = sel_idx[b * M2_ + f];
      float sc = sel_score[b * M2_ + f];
      float4 bv = boxes[(size_t)b * N_ + n];
      o[0] = (float)(c + 1); o[1] = sc;
      o[2] = bv.x; o[3] = bv.y; o[4] = bv.z; o[5] = bv.w;
    }
  }
}

// ---------------------------------------------------------------------
extern "C" void kernel_launch(void* const* d_in, const int* in_sizes, int n_in,
                              void* d_out, int out_size, void* d_ws,
                              size_t ws_size, hipStream_t stream) {
  (void)in_sizes; (void)n_in; (void)out_size; (void)ws_size;
  const float* y = (const float*)d_in[0];
  float* out = (float*)d_out;

  uint8_t* ws = (uint8_t*)d_ws;
  size_t off = 0;
  float4* boxes   = (float4*)(ws + off); off += (size_t)B_ * N_ * 16;          // 4.47 MB
  float*  scoresT = (float*)(ws + off);  off += (size_t)B_ * NCLS_ * N_ * 4;   // 22.35 MB
  float*  selS    = (float*)(ws + off);  off += (size_t)B_ * NCLS_ * NMSMAX_ * 4;
  int*    selI    = (int*)(ws + off);    off += (size_t)B_ * NCLS_ * NMSMAX_ * 4;

  decode_kernel<<<(B_ * N_ + 255) / 256, 256, 0, stream>>>(y, boxes, scoresT);
  nms_kernel<<<B_ * NCLS_, NMS_THREADS, NMS_LDS, stream>>>(boxes, scoresT,
                                                           selS, selI);
  topk_kernel<<<B_, 256, 0, stream>>>(selS, selI, boxes, out);
}